// ProjectionFunctionWrapper_42597485641914
// MI455X (gfx1250) — compile-verified
//
#include <hip/hip_runtime.h>
#include <stdint.h>

// ---------------------------------------------------------------------------
// Problem constants (from reference)
// ---------------------------------------------------------------------------
#define N_PTS   600000
#define DIMQ    192
#define NINST   50
#define NBIN    (DIMQ * DIMQ)          // 36864 bins per plane
#define PLANE_F (NBIN * NINST)         // 1,843,200 floats per p/t plane
#define EPSV    1e-8f

// d_out layout (floats), in reference return order:
//   [0*PLANE_F) xy_p  [1*PLANE_F) xz_p  [2*PLANE_F) yz_p
//   [3*PLANE_F) xy_t  [4*PLANE_F) xz_t  [5*PLANE_F) yz_t
//   [6*PLANE_F) xy_n(NBIN)  xz_n(NBIN)  yz_n(NBIN)     (written as float)
#define OUT_TOTAL (6 * PLANE_F + 3 * NBIN)   // 11,169,792

// Scatter tiling: 64 points per tile, double buffered in LDS.
#define TILE    64
#define PRED_B  (TILE * NINST * 4)     // 12800 bytes (preds), same for targets
#define CRD_B   (TILE * 4 * 4)         // 1024 bytes (4 ints per point)
#define BUF_B   (2 * PRED_B + CRD_B)   // 26624 bytes per buffer
#define NTILES  (N_PTS / TILE)         // 9375 (divides exactly)

// ---------------------------------------------------------------------------
// Kernel 1: zero the whole output (accumulators live in d_out), init mins.
// ---------------------------------------------------------------------------
__global__ void k_init(float* __restrict__ out, int* __restrict__ mins) {
  const int total4 = OUT_TOTAL / 4;
  float4 z; z.x = 0.f; z.y = 0.f; z.z = 0.f; z.w = 0.f;
  for (int i = blockIdx.x * blockDim.x + threadIdx.x; i < total4;
       i += gridDim.x * blockDim.x) {
    ((float4*)out)[i] = z;
  }
  if (blockIdx.x == 0 && threadIdx.x < 3) mins[threadIdx.x] = 0x7fffffff;
}

// ---------------------------------------------------------------------------
// Kernel 2: column mins of coords[:,1..3] (wave32 shuffle reduce + atomicMin).
// ---------------------------------------------------------------------------
__global__ void k_min(const int* __restrict__ coords, int* __restrict__ mins) {
  int m1 = 0x7fffffff, m2 = 0x7fffffff, m3 = 0x7fffffff;
  for (int i = blockIdx.x * blockDim.x + threadIdx.x; i < N_PTS;
       i += gridDim.x * blockDim.x) {
    const int* c = coords + (size_t)i * 4;
    m1 = min(m1, c[1]); m2 = min(m2, c[2]); m3 = min(m3, c[3]);
  }
  // wave32 reduction
  for (int off = 16; off > 0; off >>= 1) {
    m1 = min(m1, __shfl_xor(m1, off, 32));
    m2 = min(m2, __shfl_xor(m2, off, 32));
    m3 = min(m3, __shfl_xor(m3, off, 32));
  }
  if ((threadIdx.x & 31) == 0) {
    atomicMin(&mins[0], m1);
    atomicMin(&mins[1], m2);
    atomicMin(&mins[2], m3);
  }
}

// ---------------------------------------------------------------------------
// Async-tensor staging: issue one tile's preds/targets/coords into LDS via the
// CDNA5 async DMA path (GLOBAL_LOAD_ASYNC_TO_LDS_B128, tracked by ASYNCcnt).
// GVS addressing: uniform SGPR base per region, per-lane u32 byte offset,
// per-lane LDS byte address in VDST (workgroup-relative).
// ---------------------------------------------------------------------------
__device__ __forceinline__ void issue_async_b128(unsigned lds_addr,
                                                 unsigned goff,
                                                 const char* base) {
  asm volatile("global_load_async_to_lds_b128 %0, %1, %2 offset:0"
               :: "v"(lds_addr), "v"(goff), "s"(base)
               : "memory");
}

__device__ __forceinline__ void issue_tile(const char* preds, const char* targs,
                                           const char* coords, int tile,
                                           unsigned lds_buf, int tid, int nth) {
  // preds: 800 x 16B chunks
  for (int c = tid; c < PRED_B / 16; c += nth)
    issue_async_b128(lds_buf + (unsigned)c * 16u,
                     (unsigned)tile * (unsigned)PRED_B + (unsigned)c * 16u,
                     preds);
  // targets: 800 x 16B chunks
  for (int c = tid; c < PRED_B / 16; c += nth)
    issue_async_b128(lds_buf + (unsigned)PRED_B + (unsigned)c * 16u,
                     (unsigned)tile * (unsigned)PRED_B + (unsigned)c * 16u,
                     targs);
  // coords: 64 x 16B chunks
  for (int c = tid; c < CRD_B / 16; c += nth)
    issue_async_b128(lds_buf + 2u * (unsigned)PRED_B + (unsigned)c * 16u,
                     (unsigned)tile * (unsigned)CRD_B + (unsigned)c * 16u,
                     coords);
}

// ---------------------------------------------------------------------------
// Kernel 3: main scatter. Double-buffered async LDS staging + L2 atomics.
// Lane->channel mapping keeps each wave's 32 atomics on <=4 cache lines.
// ---------------------------------------------------------------------------
__global__ void k_scatter(const int* __restrict__ coords,
                          const float* __restrict__ preds,
                          const float* __restrict__ targs,
                          float* __restrict__ out,
                          const int* __restrict__ mins) {
  __shared__ __align__(16) unsigned char smem[2 * BUF_B];
  const int tid = threadIdx.x;
  const int nth = blockDim.x;

  // Workgroup-relative LDS byte address of smem: low 32 bits of flat address
  // (ISA 10.2, LDS aperture).
  const unsigned lds0 = (unsigned)(uintptr_t)(&smem[0]);

  const int m1 = mins[0], m2 = mins[1], m3 = mins[2];

  float* xy_p = out;
  float* xz_p = out + 1 * (size_t)PLANE_F;
  float* yz_p = out + 2 * (size_t)PLANE_F;
  float* xy_t = out + 3 * (size_t)PLANE_F;
  float* xz_t = out + 4 * (size_t)PLANE_F;
  float* yz_t = out + 5 * (size_t)PLANE_F;
  int*   cnt  = (int*)(out + 6 * (size_t)PLANE_F);   // 3*NBIN int accumulators

  const char* pB = (const char*)preds;
  const char* tB = (const char*)targs;
  const char* cB = (const char*)coords;

  int buf = 0;
  if ((int)blockIdx.x < NTILES)
    issue_tile(pB, tB, cB, blockIdx.x, lds0, tid, nth);

  for (int t = blockIdx.x; t < NTILES; t += gridDim.x) {
    // Wait for this wave's async DMA, then make all waves' LDS writes visible.
    asm volatile("s_wait_asynccnt 0x0" ::: "memory");
    __syncthreads();

    // Prefetch the next tile into the other buffer while we consume this one.
    const int tn = t + gridDim.x;
    if (tn < NTILES)
      issue_tile(pB, tB, cB, tn, lds0 + (unsigned)((buf ^ 1) * BUF_B), tid, nth);

    const float* sp = (const float*)(smem + buf * BUF_B);
    const float* st = (const float*)(smem + buf * BUF_B + PRED_B);
    const int*   sc = (const int*)  (smem + buf * BUF_B + 2 * PRED_B);

    for (int j = tid; j < TILE * NINST; j += nth) {
      const int p = j / NINST;
      const int i = j - p * NINST;
      const int x = sc[p * 4 + 1] - m1;
      const int y = sc[p * 4 + 2] - m2;
      const int z = sc[p * 4 + 3] - m3;
      const int bxy = x * DIMQ + y;
      const int bxz = x * DIMQ + z;
      const int byz = y * DIMQ + z;
      const float pv = sp[j];
      const float tv = st[j];
      atomicAdd(&xy_p[(size_t)bxy * NINST + i], pv);
      atomicAdd(&xz_p[(size_t)bxz * NINST + i], pv);
      atomicAdd(&yz_p[(size_t)byz * NINST + i], pv);
      atomicAdd(&xy_t[(size_t)bxy * NINST + i], tv);
      atomicAdd(&xz_t[(size_t)bxz * NINST + i], tv);
      atomicAdd(&yz_t[(size_t)byz * NINST + i], tv);
      if (i == 0) {
        atomicAdd(&cnt[bxy], 1);
        atomicAdd(&cnt[NBIN + bxz], 1);
        atomicAdd(&cnt[2 * NBIN + byz], 1);
      }
    }
    __syncthreads();   // buffer 'buf' free for the next iteration's prefetch
    buf ^= 1;
  }
}

// ---------------------------------------------------------------------------
// Kernel 4: normalize p and t in place by (cnt + eps), 0 for empty bins.
// Counts are still int here (converted by k_cnt afterwards).
// ---------------------------------------------------------------------------
__global__ void k_norm(float* __restrict__ out) {
  const int* cnt = (const int*)(out + 6 * (size_t)PLANE_F);
  const int total = 3 * PLANE_F;
  for (int i = blockIdx.x * blockDim.x + threadIdx.x; i < total;
       i += gridDim.x * blockDim.x) {
    const int plane = i / PLANE_F;
    const int r = i - plane * PLANE_F;
    const int bin = r / NINST;
    const int c = cnt[plane * NBIN + bin];
    const float inv = (c > 0) ? 1.0f / ((float)c + EPSV) : 0.0f;
    out[(size_t)plane * PLANE_F + r]       *= inv;   // p plane
    out[(size_t)(plane + 3) * PLANE_F + r] *= inv;   // t plane
  }
}

// ---------------------------------------------------------------------------
// Kernel 5: convert counts int -> float in place (runs after k_norm).
// ---------------------------------------------------------------------------
__global__ void k_cnt(float* __restrict__ out) {
  int* cnt = (int*)(out + 6 * (size_t)PLANE_F);
  const int total = 3 * NBIN;
  for (int i = blockIdx.x * blockDim.x + threadIdx.x; i < total;
       i += gridDim.x * blockDim.x) {
    const int c = cnt[i];
    ((float*)cnt)[i] = (float)c;
  }
}

// ---------------------------------------------------------------------------
// Launch
// ---------------------------------------------------------------------------
extern "C" void kernel_launch(void* const* d_in, const int* in_sizes, int n_in,
                              void* d_out, int out_size, void* d_ws, size_t ws_size,
                              hipStream_t stream) {
  const int*   coords = (const int*)d_in[0];
  const float* preds  = (const float*)d_in[1];
  const float* targs  = (const float*)d_in[2];
  float* out = (float*)d_out;
  int*   mins = (int*)d_ws;           // 3 ints of scratch

  k_init   <<<1024, 256, 0, stream>>>(out, mins);
  k_min    <<< 256, 256, 0, stream>>>(coords, mins);
  k_scatter<<<2048, 256, 0, stream>>>(coords, preds, targs, out, mins);
  k_norm   <<<4096, 256, 0, stream>>>(out);
  k_cnt    <<< 128, 256, 0, stream>>>(out);
}